// EulerRosenbrockModel_53180285059254
// MI455X (gfx1250) — compile-verified
//
#include <hip/hip_runtime.h>
#include <math.h>

// ---------------------------------------------------------------------------
// EulerRosenbrock exponential-integrator step, reformulated as a Neumann
// series so the per-row 256x256 Jacobian solve collapses into batched GEMMs:
//   J*X = W2 (S .* (X W1^T))     (never form J)
//   out = sum_{k=0..6} B^k rhs,  B = (H/3) J,  rhs = V + (H/6) J V
// All GEMMs run on V_WMMA_F32_16X16X4_F32 (f32 in / f32 out, matches ref).
// Each wave register-blocks a 16x64 strip (A fragment reused across 4 WMMAs),
// with explicit double-buffered fragment prefetch so loads for step k+4 are
// in flight while the 4 WMMAs for step k execute. N and K are template
// constants so all B-fragment addresses are immediate offsets.
// ---------------------------------------------------------------------------

typedef float v2f __attribute__((ext_vector_type(2)));
typedef float v8f __attribute__((ext_vector_type(8)));

#define H_STEP 0.01f

// LDS-tiled transpose: coalesced loads and stores (rows, cols multiples of 32).
__global__ __launch_bounds__(256) void transpose_f32(
    const float* __restrict__ in, float* __restrict__ out, int rows, int cols) {
    __shared__ float tile[32][33];
    const int cb = blockIdx.x * 32;   // col base (input)
    const int rb = blockIdx.y * 32;   // row base (input)
#pragma unroll
    for (int j = 0; j < 32; j += 8)
        tile[threadIdx.y + j][threadIdx.x] =
            in[(size_t)(rb + threadIdx.y + j) * cols + cb + threadIdx.x];
    __syncthreads();
#pragma unroll
    for (int j = 0; j < 32; j += 8)
        out[(size_t)(cb + threadIdx.y + j) * rows + rb + threadIdx.x] =
            tile[threadIdx.x][threadIdx.y + j];
}

// One wave computes a 16x64 tile strip of C = A(MxK) * B(KxN), row-major,
// K stepped by 4 via V_WMMA_F32_16X16X4_F32. Epilogue fused by MODE:
//   0: t = tanh(p + bias[n]); out0 = t; out1 = 1 - t*t          (H stage)
//   1: out0 = p + bias[n]                                        (V stage)
//   2: out0 = p * emat[idx]                                      (S-mask stage)
//   3: r = emat[idx] + alpha*p; out0 = r; out1 = r               (rhs init)
//   4: r = alpha*p;            out0 = r; out1 += r               (Neumann accum)
template <int MODE, int N, int K>
__global__ __launch_bounds__(128) void gemm16x64_wmma(
    const float* __restrict__ A, const float* __restrict__ B,
    const float* __restrict__ bias, const float* __restrict__ emat,
    float* __restrict__ out0, float* __restrict__ out1, float alpha)
{
    const int lane = threadIdx.x;            // 0..31, full wave, no divergence
    const int m0   = (blockIdx.y * 4 + threadIdx.y) * 16;
    const int n0   = blockIdx.x * 64;

    const int ml   = lane & 15;              // matrix row (A) / col (B,C)
    const int koff = (lane >> 4) * 2;        // K sub-offset per ISA layout

    v8f acc[4] = {};

    // A fragment: lane l -> A[m0+l%16][k + 2*(l/16) + {0,1}] (one b64 load)
    const float* __restrict__ aptr = A + (size_t)(m0 + ml) * K + koff;
    // B fragment: lane l -> B[k + 2*(l/16) + {0,1}][n0 + 16t + l%16]
    const float* __restrict__ bptr = B + (size_t)koff * N + n0 + ml;

    // ---- prologue: fragments for k = 0 ----
    v2f a_c = *reinterpret_cast<const v2f*>(aptr);
    v2f b_c[4];
#pragma unroll
    for (int t = 0; t < 4; ++t) {
        b_c[t].x = bptr[t * 16];
        b_c[t].y = bptr[N + t * 16];
    }

    // ---- steady state: prefetch k+4 into distinct regs, then 4 WMMAs on k --
#pragma unroll 2
    for (int k = 0; k < K - 4; k += 4) {
        const float* an = aptr + (k + 4);
        const float* bn = bptr + (size_t)(k + 4) * N;
        v2f a_n = *reinterpret_cast<const v2f*>(an);
        v2f b_n[4];
#pragma unroll
        for (int t = 0; t < 4; ++t) {
            b_n[t].x = bn[t * 16];
            b_n[t].y = bn[N + t * 16];
        }
#pragma unroll
        for (int t = 0; t < 4; ++t)
            acc[t] = __builtin_amdgcn_wmma_f32_16x16x4_f32(
                         false, a_c, false, b_c[t], (short)0, acc[t], false, false);
        a_c = a_n;
#pragma unroll
        for (int t = 0; t < 4; ++t) b_c[t] = b_n[t];
    }

    // ---- epilogue WMMAs for the last K-step ----
#pragma unroll
    for (int t = 0; t < 4; ++t)
        acc[t] = __builtin_amdgcn_wmma_f32_16x16x4_f32(
                     false, a_c, false, b_c[t], (short)0, acc[t], false, false);

    // C/D layout: VGPR r holds row m0 + r + 8*(lane/16), col n + lane%16
    const int mbase = m0 + (lane >> 4) * 8;
#pragma unroll
    for (int t = 0; t < 4; ++t) {
        const int nc = n0 + t * 16 + ml;
#pragma unroll
        for (int r = 0; r < 8; ++r) {
            const size_t idx = (size_t)(mbase + r) * N + nc;
            const float p = acc[t][r];
            if constexpr (MODE == 0) {
                float th = tanhf(p + bias[nc]);
                out0[idx] = th;
                out1[idx] = 1.0f - th * th;
            } else if constexpr (MODE == 1) {
                out0[idx] = p + bias[nc];
            } else if constexpr (MODE == 2) {
                out0[idx] = p * emat[idx];
            } else if constexpr (MODE == 3) {
                float rr = emat[idx] + alpha * p;
                out0[idx] = rr;
                out1[idx] = rr;
            } else {
                float rr = alpha * p;
                out0[idx] = rr;
                out1[idx] += rr;
            }
        }
    }
}

extern "C" void kernel_launch(void* const* d_in, const int* in_sizes, int n_in,
                              void* d_out, int out_size, void* d_ws, size_t ws_size,
                              hipStream_t stream) {
    constexpr int BATCH = 512;
    constexpr int DIM   = 256;
    constexpr int HID   = 1024;

    const float* Y  = (const float*)d_in[0];   // (512, 256)
    const float* W1 = (const float*)d_in[1];   // (1024, 256)
    const float* b1 = (const float*)d_in[2];   // (1024,)
    const float* W2 = (const float*)d_in[3];   // (256, 1024)
    const float* b2 = (const float*)d_in[4];   // (256,)
    float* OUT = (float*)d_out;                // (512, 256) accumulator lives here

    // Workspace carve-up (floats): ~9.4 MB total
    float* w = (float*)d_ws;
    float* W1T = w; w += (size_t)DIM * HID;    // (256,1024) = W1^T
    float* W2T = w; w += (size_t)HID * DIM;    // (1024,256) = W2^T
    float* T   = w; w += (size_t)BATCH * HID;  // tanh(h)
    float* S   = w; w += (size_t)BATCH * HID;  // 1 - tanh^2
    float* V   = w; w += (size_t)BATCH * DIM;  // f(y)
    float* Zs  = w; w += (size_t)BATCH * HID;  // S .* (X W1^T)
    float* X   = w; w += (size_t)BATCH * DIM;  // Neumann iterate

    // 1) Transpose weights once so every GEMM B-load is N-coalesced.
    {
        const dim3 tb(32, 8);
        transpose_f32<<<dim3(DIM / 32, HID / 32), tb, 0, stream>>>(W1, W1T, HID, DIM);
        transpose_f32<<<dim3(HID / 32, DIM / 32), tb, 0, stream>>>(W2, W2T, DIM, HID);
    }

    const dim3 blk(32, 4);
    const dim3 grdWide(HID / 64, BATCH / 64);   // N=1024 GEMMs
    const dim3 grdNarrow(DIM / 64, BATCH / 64); // N=256 GEMMs

    // 2) H stage: T = tanh(Y W1^T + b1), S = 1 - T^2          (512x1024, K=256)
    gemm16x64_wmma<0, HID, DIM><<<grdWide, blk, 0, stream>>>(
        Y, W1T, b1, nullptr, T, S, 0.0f);

    // 3) V stage: V = T W2^T + b2                              (512x256, K=1024)
    gemm16x64_wmma<1, DIM, HID><<<grdNarrow, blk, 0, stream>>>(
        T, W2T, b2, nullptr, V, nullptr, 0.0f);

    // 4) J*V, first half: Zs = S .* (V W1^T)
    gemm16x64_wmma<2, HID, DIM><<<grdWide, blk, 0, stream>>>(
        V, W1T, nullptr, S, Zs, nullptr, 0.0f);

    // 5) rhs = V + (H/6) * (Zs W2^T);  X = rhs;  OUT(acc) = rhs
    gemm16x64_wmma<3, DIM, HID><<<grdNarrow, blk, 0, stream>>>(
        Zs, W2T, nullptr, V, X, OUT, H_STEP / 6.0f);

    // 6) Neumann series: X <- (H/3) J X; OUT += X. ||(H/3)J||~0.015 -> 6 terms
    //    gives ~1e-13 truncation, below f32 roundoff.
    for (int it = 0; it < 6; ++it) {
        gemm16x64_wmma<2, HID, DIM><<<grdWide, blk, 0, stream>>>(
            X, W1T, nullptr, S, Zs, nullptr, 0.0f);
        gemm16x64_wmma<4, DIM, HID><<<grdNarrow, blk, 0, stream>>>(
            Zs, W2T, nullptr, nullptr, X, OUT, H_STEP / 3.0f);
    }
}